// GNNLayer_78872779423829
// MI455X (gfx1250) — compile-verified
//
#include <hip/hip_runtime.h>
#include <hip/hip_bf16.h>
#include <math.h>

#define N_NODES 50000
#define NUM_EDGES 800000
#define NH 4
#define D_IN 256
#define D_OUT 64
#define NCOL (NH * D_OUT) /* 256 */

typedef __attribute__((ext_vector_type(2))) float v2f;
typedef __attribute__((ext_vector_type(8))) float v8f;

// ---------------------------------------------------------------------------
// Kernel 1: feats[h][n][o] = x[n][:] @ Ws[h][:][o] + bs[h][o]
// One block (8 wave32) per 16-row M-tile. A staged in LDS (pitch 260 floats:
// breaks the k%64 bank-conflict pattern, keeps float4/b64 alignment).
// Each wave computes TWO adjacent 16x16 output tiles of the 256-wide (H*O)
// output in a single fused K-loop with two independent accumulators:
//   - one shared ds_load for the A fragment per K-step
//   - two independent v_wmma_f32_16x16x4_f32 chains (hides XDL latency)
// ---------------------------------------------------------------------------
__global__ __launch_bounds__(256) void gat_feats_wmma(
    const float* __restrict__ x,     // [N, 256]
    const float* __restrict__ Ws,    // [H, 256, 64]
    const float* __restrict__ bs,    // [H, 64]
    float* __restrict__ feats)       // [H, N, 64]
{
    __shared__ float As[16 * 260];   // 16 rows x 256 K, pitch 260 floats
    const int mt = blockIdx.x;       // 0..3124  (50000/16)
    const int t  = threadIdx.x;

    // Cooperative stage of A: 16*256 floats = 1024 float4, 4 per thread.
    {
        const float4* xg = (const float4*)(x + (size_t)mt * 16 * D_IN);
        #pragma unroll
        for (int j = 0; j < 4; ++j) {
            int idx4 = t + j * 256;          // 0..1023
            int row  = idx4 >> 6;            // 64 float4 per row
            int col4 = idx4 & 63;
            float4 v = xg[idx4];
            *(float4*)(&As[row * 260 + col4 * 4]) = v;
        }
    }
    __syncthreads();

    const int wave  = t >> 5;
    const int lane  = t & 31;
    const int l15   = lane & 15;
    const int khalf = (lane >> 4) << 1;      // 0 or 2 (A/B K sub-lane split)

    // Wave owns column tiles nt0 = 2*wave and nt0+1 (same head h).
    const int nt0 = wave * 2;                // 0,2,..,14
    const int h   = nt0 >> 2;
    const int o0  = (nt0 & 3) * 16;          // 0 or 32
    const float* bA = Ws + (size_t)h * D_IN * D_OUT + o0 + l15;
    const float* bB = bA + 16;

    v8f acc0 = {};
    v8f acc1 = {};
    #pragma unroll 8
    for (int k0 = 0; k0 < D_IN; k0 += 4) {
        const int kb = k0 + khalf;
        // A 16x4 f32: lane = M (0..15), VGPR j holds K = khalf + j (shared)
        v2f a = *(const v2f*)(&As[l15 * 260 + kb]);
        // B 4x16 f32: lane = N, VGPR j holds K = khalf + j
        v2f b0, b1;
        b0.x = bA[(size_t)kb * D_OUT];
        b0.y = bA[(size_t)(kb + 1) * D_OUT];
        b1.x = bB[(size_t)kb * D_OUT];
        b1.y = bB[(size_t)(kb + 1) * D_OUT];
        acc0 = __builtin_amdgcn_wmma_f32_16x16x4_f32(
                   false, a, false, b0, (short)0, acc0, false, false);
        acc1 = __builtin_amdgcn_wmma_f32_16x16x4_f32(
                   false, a, false, b1, (short)0, acc1, false, false);
    }

    // C/D layout: VGPR r holds M=r (lanes 0-15) / M=r+8 (lanes 16-31)
    const float bias0 = bs[h * D_OUT + o0 + l15];
    const float bias1 = bs[h * D_OUT + o0 + 16 + l15];
    float* outp = feats + ((size_t)h * N_NODES + mt * 16) * D_OUT + o0 + l15;
    const int mbase = (lane >> 4) << 3;
    #pragma unroll
    for (int r = 0; r < 8; ++r) {
        outp[(size_t)(r + mbase) * D_OUT]      = acc0[r] + bias0;
        outp[(size_t)(r + mbase) * D_OUT + 16] = acc1[r] + bias1;
    }
}

// ---------------------------------------------------------------------------
// Kernel 2: per-(h,n) attention logits  a_s = feats . aw_src, a_d = feats . aw_dst
// One wave per (h,n); 2 elements per lane + shfl tree reduction.
// ---------------------------------------------------------------------------
__global__ __launch_bounds__(256) void gat_attn_scores(
    const float* __restrict__ feats,  // [H*N, 64]
    const float* __restrict__ aw,     // [H, 129]
    float* __restrict__ a_s,          // [H*N]
    float* __restrict__ a_d)          // [H*N]
{
    int gid = blockIdx.x * 8 + (threadIdx.x >> 5);   // gid = h*N + n
    if (gid >= NH * N_NODES) return;
    int h    = gid / N_NODES;
    int lane = threadIdx.x & 31;
    const float* f   = feats + (size_t)gid * D_OUT;
    const float* awh = aw + h * (2 * D_OUT + 1);
    float f0 = f[lane], f1 = f[lane + 32];
    float ps = f0 * awh[lane]      + f1 * awh[lane + 32];
    float pd = f0 * awh[64 + lane] + f1 * awh[96 + lane];
    #pragma unroll
    for (int off = 16; off > 0; off >>= 1) {
        ps += __shfl_down(ps, off, 32);
        pd += __shfl_down(pd, off, 32);
    }
    if (lane == 0) { a_s[gid] = ps; a_d[gid] = pd; }
}

// ---------------------------------------------------------------------------
// Kernel 3: edge scatter. One wave per edge. Lanes 0..3 compute per-head
// weight w[h] = exp(-relu(score/20)) and accumulate row_sum; all lanes
// scatter w*feats[dst] into pooled (== d_out, layout [n, h*64+o]).
// feats + pooled are L2-resident (51 MB each, 192 MB L2) -> runs at L2 /
// atomic-unit throughput, not HBM.
// ---------------------------------------------------------------------------
__global__ __launch_bounds__(256) void gat_edge_scatter(
    const float* __restrict__ feats,  // [H, N, 64]
    const int*   __restrict__ idx,    // [2, E]
    const float* __restrict__ elem,   // [E]
    const float* __restrict__ aw,     // [H, 129]
    const float* __restrict__ ab,     // [H]
    const float* __restrict__ a_s,    // [H*N]
    const float* __restrict__ a_d,    // [H*N]
    float* __restrict__ pooled,       // d_out: [N, 256]
    float* __restrict__ row_sum)      // [H, N]
{
    int e = blockIdx.x * 8 + (threadIdx.x >> 5);
    if (e >= NUM_EDGES) return;
    int lane = threadIdx.x & 31;
    int src  = idx[e];
    int dst  = idx[NUM_EDGES + e];

    float w = 0.f;
    if (lane < NH) {
        int h = lane;
        float score = a_s[h * N_NODES + src] + a_d[h * N_NODES + dst]
                    + aw[h * 129 + 128] * elem[e] + ab[h];
        float rel = score * 0.05f;            // score / 20
        rel = rel > 0.f ? rel : 0.f;
        w = __expf(-rel);
        atomicAdd(&row_sum[h * N_NODES + src], w);
    }
    #pragma unroll
    for (int h = 0; h < NH; ++h) {
        float wh = __shfl(w, h, 32);
        const float* fd = feats + ((size_t)h * N_NODES + dst) * D_OUT;
        float* pp = pooled + (size_t)src * NCOL + h * D_OUT;
        atomicAdd(&pp[lane],      wh * fd[lane]);
        atomicAdd(&pp[lane + 32], wh * fd[lane + 32]);
    }
}

// ---------------------------------------------------------------------------
// Kernel 4: out[n, h*64+o] /= row_sum[h, n]
// ---------------------------------------------------------------------------
__global__ __launch_bounds__(256) void gat_finalize(
    float* __restrict__ out, const float* __restrict__ row_sum)
{
    size_t tid = (size_t)blockIdx.x * 256 + threadIdx.x;
    if (tid >= (size_t)N_NODES * NCOL) return;
    int n = (int)(tid >> 8);
    int c = (int)(tid & 255);
    int h = c >> 6;
    out[tid] = out[tid] / row_sum[h * N_NODES + n];
}

// ---------------------------------------------------------------------------
extern "C" void kernel_launch(void* const* d_in, const int* in_sizes, int n_in,
                              void* d_out, int out_size, void* d_ws, size_t ws_size,
                              hipStream_t stream)
{
    const float* x    = (const float*)d_in[0];
    const int*   idx  = (const int*)  d_in[1];
    const float* elem = (const float*)d_in[2];
    const float* Ws   = (const float*)d_in[3];
    const float* bs   = (const float*)d_in[4];
    const float* aw   = (const float*)d_in[5];
    const float* ab   = (const float*)d_in[6];
    float* out = (float*)d_out;

    // Workspace layout: feats | a_s | a_d | row_sum  (~53.6 MB)
    float* feats   = (float*)d_ws;                              // H*N*64
    float* a_s     = feats + (size_t)NH * N_NODES * D_OUT;      // H*N
    float* a_d     = a_s   + (size_t)NH * N_NODES;              // H*N
    float* row_sum = a_d   + (size_t)NH * N_NODES;              // H*N

    // Accumulators must be zeroed every call (graph-capture-safe async memset).
    hipMemsetAsync(d_out, 0, (size_t)out_size * sizeof(float), stream);
    hipMemsetAsync(row_sum, 0, (size_t)NH * N_NODES * sizeof(float), stream);

    gat_feats_wmma<<<N_NODES / 16, 256, 0, stream>>>(x, Ws, bs, feats);
    gat_attn_scores<<<(NH * N_NODES + 7) / 8, 256, 0, stream>>>(feats, aw, a_s, a_d);
    gat_edge_scatter<<<NUM_EDGES / 8, 256, 0, stream>>>(feats, idx, elem, aw, ab,
                                                        a_s, a_d, out, row_sum);
    gat_finalize<<<((size_t)N_NODES * NCOL + 255) / 256, 256, 0, stream>>>(out, row_sum);
}